// UMPS_35390530519710
// MI455X (gfx1250) — compile-verified
//
#include <hip/hip_runtime.h>

// ---- problem constants (match reference) ----
#define BOND   64
#define FEAT1  33            // FEAT + flag channel
#define BATCH  256
#define TLEN   512
#define OUTD   32
#define NWAVE  16
#define KCHUNK 4             // K reduction split across 4 wave groups (by f % 4)
#define VSTRIDE 66           // padded LDS stride for v (bank-conflict free)

typedef __attribute__((ext_vector_type(2))) float v2f;
typedef __attribute__((ext_vector_type(4))) float v4f;
typedef __attribute__((ext_vector_type(8))) float v8f;

// Repack core (i, f, j) -> (f*64 + i, j) row-major so each feature slice is a
// contiguous 64x64 block: K-blocks of 4 never straddle f, and all offsets
// inside a feature slice are compile-time immediates.
__global__ __launch_bounds__(256) void umps_permute_core(
        const float* __restrict__ core, float* __restrict__ corePerm) {
    int idx = blockIdx.x * 256 + threadIdx.x;
    if (idx < BOND * FEAT1 * BOND) {
        int j    = idx & 63;
        int rest = idx >> 6;
        int f    = rest % FEAT1;
        int i    = rest / FEAT1;
        corePerm[(f * BOND + i) * BOND + j] = core[idx];
    }
}

// One workgroup per 16-batch tile. 16 waves:
//   wave w -> col-tile c = w&3 (16 bond columns), K-chunk q = w>>2
//   chunk q covers features f = q, q+4, q+8, ... (< 33)
// Per timestep: v_{t+1} = (x_t (x) v_t) @ corePerm via V_WMMA_F32_16X16X4_F32.
__global__ __launch_bounds__(512) void umps_chain_kernel(
        const float* __restrict__ xg,        // (B, T, 33)
        const float* __restrict__ corePerm,  // (2112, 64) row-major, f-major
        const float* __restrict__ alpha,     // (64)
        const float* __restrict__ outCore,   // (64, 32)
        float* __restrict__ out) {           // (B, 32)
    __shared__ float v_lds[16 * VSTRIDE];    // carry, padded rows
    __shared__ float x_lds[16 * FEAT1];      // current timestep features
    __shared__ float part[NWAVE * 256];      // per-wave partial C tiles

    const int tid   = threadIdx.x;
    const int w     = tid >> 5;      // wave id 0..15
    const int lane  = tid & 31;
    const int h     = lane >> 4;     // wave half (K parity pair select)
    const int mn    = lane & 15;     // M for A / N for B
    const int c     = w & 3;         // column tile
    const int q     = w >> 2;        // K chunk = feature residue class
    const int jBase = c << 4;
    const int bBase = blockIdx.x * 16;

    // v[r][i] = alpha[i] for all rows of this batch tile
    #pragma unroll
    for (int u = 0; u < 2; ++u) {
        int e = tid + u * 512;
        v_lds[(e >> 6) * VSTRIDE + (e & 63)] = alpha[e & 63];
    }
    // stage x for t = 0
    if (tid < 16 * FEAT1) {
        int r = tid / FEAT1, f = tid % FEAT1;
        x_lds[tid] = xg[((size_t)(bBase + r) * TLEN + 0) * FEAT1 + f];
    }
    __syncthreads();

    // Fixed per-lane pointers. Assumed B (4x16 f32) lane layout mirrors A/C:
    // lane holds K = 2h, 2h+1 within each K-block of 4.
    const float* xrow = &x_lds[mn * FEAT1];
    const float* vp   = &v_lds[mn * VSTRIDE + 2 * h];
    const float* bq   = corePerm + ((size_t)q * BOND + 2 * h) * BOND + jBase + mn;

    for (int t = 0; t < TLEN; ++t) {
        // hint next timestep's x toward the caches (global_prefetch_b8)
        if (t + 1 < TLEN && tid < 16 * FEAT1) {
            int r = tid / FEAT1, f = tid % FEAT1;
            __builtin_prefetch(
                &xg[((size_t)(bBase + r) * TLEN + (t + 1)) * FEAT1 + f], 0, 0);
        }

        v8f acc = {};
        for (int f = q; f < FEAT1; f += KCHUNK) {
            float xv = xrow[f];                               // 1 ds_load_b32 / 16 WMMA
            const float* bp = bq + (size_t)(f - q) * BOND * BOND;
            #pragma unroll
            for (int ib = 0; ib < 16; ++ib) {                 // i = 4*ib .. 4*ib+3
                v2f vv = *(const v2f*)(vp + 4 * ib);          // ds_load_b64 offset:imm
                v2f a; a.x = vv.x * xv; a.y = vv.y * xv;      // v_pk_mul_f32
                v2f b;                                        // 2x global_load_b32 imm
                b.x = bp[ib * 4 * BOND];
                b.y = bp[ib * 4 * BOND + BOND];
                acc = __builtin_amdgcn_wmma_f32_16x16x4_f32(
                    false, a, false, b, (short)0, acc, false, false);
            }
        }

        // stash partial 16x16 C tile: part[((w*2+h)*16 + n)*8 + p]
        float* pst = &part[((((w << 1) | h) * 16 + mn) << 3)];
        v4f lo = { acc.s0, acc.s1, acc.s2, acc.s3 };
        v4f hi = { acc.s4, acc.s5, acc.s6, acc.s7 };
        *(v4f*)(pst)     = lo;   // ds_store_b128
        *(v4f*)(pst + 4) = hi;
        __syncthreads();

        // reduce the 4 K-chunk partials -> new carry v (exactly 2 elems/thread)
        #pragma unroll
        for (int u = 0; u < 2; ++u) {
            int e = tid + u * 512;
            int r = e >> 6, j = e & 63;
            int ct = j >> 4, n = j & 15, hh = r >> 3, p = r & 7;
            float s = 0.f;
            #pragma unroll
            for (int qq = 0; qq < KCHUNK; ++qq)
                s += part[((((qq * 4 + ct) << 1) | hh) * 16 + n) * 8 + p];
            v_lds[r * VSTRIDE + j] = s;
        }
        // stage x for t+1 (old x fully consumed before the barrier above)
        if (t + 1 < TLEN && tid < 16 * FEAT1) {
            int r = tid / FEAT1, f = tid % FEAT1;
            x_lds[tid] = xg[((size_t)(bBase + r) * TLEN + (t + 1)) * FEAT1 + f];
        }
        __syncthreads();
    }

    // out = v @ output_core : (16 x 64) @ (64 x 32) -- negligible, plain FMA
    if (tid < 16 * OUTD) {
        int r = tid >> 5, o = tid & 31;
        float s = 0.f;
        #pragma unroll 8
        for (int i = 0; i < BOND; ++i)
            s += v_lds[r * VSTRIDE + i] * outCore[i * OUTD + o];
        out[(size_t)(bBase + r) * OUTD + o] = s;
    }
}

extern "C" void kernel_launch(void* const* d_in, const int* in_sizes, int n_in,
                              void* d_out, int out_size, void* d_ws, size_t ws_size,
                              hipStream_t stream) {
    const float* x     = (const float*)d_in[0];  // (256, 512, 33)
    const float* core  = (const float*)d_in[1];  // (64, 33, 64)
    const float* alpha = (const float*)d_in[2];  // (64)
    const float* outc  = (const float*)d_in[3];  // (64, 32)
    float* out = (float*)d_out;                  // (256, 32)
    float* corePerm = (float*)d_ws;              // 2112 * 64 floats = 540 KB

    umps_permute_core<<<(BOND * FEAT1 * BOND + 255) / 256, 256, 0, stream>>>(
        core, corePerm);
    umps_chain_kernel<<<BATCH / 16, 512, 0, stream>>>(
        x, corePerm, alpha, outc, out);
}